// Mamba2Layer_54511724920851
// MI455X (gfx1250) — compile-verified
//
#include <hip/hip_runtime.h>
#include <hip/hip_bf16.h>

typedef __attribute__((ext_vector_type(2))) float v2f;
typedef __attribute__((ext_vector_type(8))) float v8f;
typedef __attribute__((ext_vector_type(4))) unsigned int u32x4;
typedef __attribute__((ext_vector_type(8))) unsigned int u32x8;

#define D_MODEL   1024
#define D_STATE   128
#define D_CONV    4
#define HEADDIM   64
#define CHUNK     64
#define D_INNER   2048
#define NHEADS    32
#define CONV_DIM  2304
#define D_IN_PROJ 4384
#define BATCH     2
#define SEQ       4096
#define NCHUNK    (SEQ / CHUNK)
#define BL        (BATCH * SEQ)

__device__ __forceinline__ v8f wmma_f32(v2f a, v2f b, v8f c) {
  return __builtin_amdgcn_wmma_f32_16x16x4_f32(false, a, false, b, (short)0, c, false, false);
}

__device__ __forceinline__ float sigmoidf_(float x) { return 1.0f / (1.0f + __expf(-x)); }

// ---------------------------------------------------------------------------
// TDM: async 2D fp32 tile load Global -> LDS (CDNA5 ISA sec.8 descriptor).
// tile_w = dwords per row, tile_h = rows, row_stride in elements.
// pad_i: pad interval code (2^(code+1) dwords), pad_a: pad amount code (code+1 dwords)
// ---------------------------------------------------------------------------
__device__ __forceinline__ void tdm_load_2d(void* lds_ptr, const float* gptr,
                                            unsigned tile_w, unsigned tile_h,
                                            unsigned long long row_stride,
                                            unsigned pad_i, unsigned pad_a) {
  const unsigned long long ga = (unsigned long long)(size_t)gptr;
  const unsigned lds_addr = (unsigned)(size_t)lds_ptr;   // low 32 bits = LDS offset
  u32x4 g0;
  g0[0] = 1u;                                            // count=1 (valid user D#)
  g0[1] = lds_addr;
  g0[2] = (unsigned)(ga & 0xFFFFFFFFu);                  // global_addr[31:0]
  g0[3] = (unsigned)(ga >> 32) | 0x80000000u;            // global_addr[56:32] | type=2
  u32x8 g1;
  g1[0] = (2u << 16)                                     // data_size = 4B
        | (1u << 20)                                     // pad_enable
        | (pad_i << 22) | (pad_a << 25);
  g1[1] = (tile_w & 0xFFFFu) << 16;                      // tensor_dim0[15:0]
  g1[2] = ((tile_w >> 16) & 0xFFFFu) | ((tile_h & 0xFFFFu) << 16);  // td0 hi | td1 lo
  g1[3] = ((tile_h >> 16) & 0xFFFFu) | (tile_w << 16);   // td1 hi | tile_dim0
  g1[4] = tile_h & 0xFFFFu;                              // tile_dim1 | tile_dim2=0
  g1[5] = (unsigned)(row_stride & 0xFFFFFFFFu);          // dim0 stride lo
  g1[6] = (unsigned)(row_stride >> 32) & 0xFFFFu;        // dim0 stride hi | dim1 stride=0
  g1[7] = 0u;
  const u32x4 gz = (u32x4){0u, 0u, 0u, 0u};              // groups 2/3 unused (2D)
  asm volatile("tensor_load_to_lds %0, %1, %2, %3"
               :: "s"(g0), "s"(g1), "s"(gz), "s"(gz)
               : "memory");
}

// ---------------------------------------------------------------------------
// fp32 GEMM: C[M,N] = A[M,K] * W[K,N].  256 thr (8 waves).
// Block tile 256(M) x 32(N), K staged 32 at a time in LDS via TDM.
// Each wave: 32x32 slab => 2x2 register blocking, 4 WMMA accumulators.
// ---------------------------------------------------------------------------
__global__ void gemm_f32(const float* __restrict__ A, const float* __restrict__ W,
                         float* __restrict__ C, int M, int N, int K) {
  __shared__ float As[256][33];
  __shared__ float Bs[32][33];

  const int tid  = threadIdx.x;
  const int lane = tid & 31;
  const int wv   = tid >> 5;
  const int lm   = lane & 15;
  const int hi   = lane >> 4;

  const int m0 = blockIdx.y * 256;
  const int n0 = blockIdx.x * 32;

  v8f a00 = {0.f,0.f,0.f,0.f,0.f,0.f,0.f,0.f};
  v8f a01 = a00, a10 = a00, a11 = a00;

  for (int kt = 0; kt < K; kt += 32) {
    if (wv == 0) {
      tdm_load_2d(&As[0][0], &A[(size_t)m0 * K + kt], 32, 256, (unsigned long long)K, 4, 0);
      tdm_load_2d(&Bs[0][0], &W[(size_t)kt * N + n0], 32, 32,  (unsigned long long)N, 4, 0);
      __builtin_amdgcn_s_wait_tensorcnt(0);
    }
    __syncthreads();

    #pragma unroll
    for (int kk = 0; kk < 8; ++kk) {
      const int k = kk * 4 + hi * 2;
      v2f fa0, fa1, fb0, fb1;
      fa0[0] = As[wv * 32 + lm][k];
      fa0[1] = As[wv * 32 + lm][k + 1];
      fa1[0] = As[wv * 32 + 16 + lm][k];
      fa1[1] = As[wv * 32 + 16 + lm][k + 1];
      fb0[0] = Bs[k][lm];
      fb0[1] = Bs[k + 1][lm];
      fb1[0] = Bs[k][16 + lm];
      fb1[1] = Bs[k + 1][16 + lm];
      a00 = wmma_f32(fa0, fb0, a00);
      a01 = wmma_f32(fa0, fb1, a01);
      a10 = wmma_f32(fa1, fb0, a10);
      a11 = wmma_f32(fa1, fb1, a11);
    }
    __syncthreads();
  }

  #pragma unroll
  for (int v = 0; v < 8; ++v) {
    const int m  = m0 + wv * 32 + v + 8 * hi;
    const int m2 = m + 16;
    C[(size_t)m  * N + n0 + lm]      = a00[v];
    C[(size_t)m  * N + n0 + 16 + lm] = a01[v];
    C[(size_t)m2 * N + n0 + lm]      = a10[v];
    C[(size_t)m2 * N + n0 + 16 + lm] = a11[v];
  }
}

// ---------------------------------------------------------------------------
__global__ void dt_kernel(const float* __restrict__ zx, const float* __restrict__ dt_bias,
                          const float* __restrict__ A_log,
                          float* __restrict__ dt, float* __restrict__ dA) {
  const int idx = blockIdx.x * blockDim.x + threadIdx.x;
  if (idx >= BL * NHEADS) return;
  const int h = idx & (NHEADS - 1);
  const int bl = idx >> 5;
  const float raw = zx[(size_t)bl * D_IN_PROJ + (D_IN_PROJ - NHEADS) + h] + dt_bias[h];
  const float dtv = (raw > 20.f) ? raw : log1pf(__expf(raw));
  dt[idx] = dtv;
  dA[idx] = -__expf(A_log[h]) * dtv;
}

// ---------------------------------------------------------------------------
__global__ void conv_kernel(const float* __restrict__ zx, const float* __restrict__ cw,
                            const float* __restrict__ cb, float* __restrict__ xconv) {
  const int idx = blockIdx.x * blockDim.x + threadIdx.x;
  if (idx >= BL * CONV_DIM) return;
  const int ch = idx % CONV_DIM;
  const int bl = idx / CONV_DIM;
  const int l  = bl & (SEQ - 1);
  float acc = cb[ch];
  #pragma unroll
  for (int i = 0; i < D_CONV; ++i) {
    const int li = l + i - (D_CONV - 1);
    if (li >= 0)
      acc += zx[(size_t)(bl - l + li) * D_IN_PROJ + D_INNER + ch] * cw[ch * D_CONV + i];
  }
  xconv[(size_t)bl * CONV_DIM + ch] = acc * sigmoidf_(acc);
}

// ---------------------------------------------------------------------------
// Per-(b,chunk,head): Y_diag + chunk states via fp32 WMMA, tiles staged by TDM.
// grid = (NHEADS, NCHUNK, BATCH), block = 256
// ---------------------------------------------------------------------------
__global__ void chunk_kernel(const float* __restrict__ xconv, const float* __restrict__ dt,
                             const float* __restrict__ dA,
                             float* __restrict__ ydiag, float* __restrict__ states,
                             float* __restrict__ dAsum) {
  __shared__ float Bsm[CHUNK][130];
  __shared__ float Csm[CHUNK][130];
  __shared__ float Xs[CHUNK][66];
  __shared__ float Gs[CHUNK][66];
  __shared__ float Acs[CHUNK];
  __shared__ float dts[CHUNK];

  const int tid  = threadIdx.x;
  const int lane = tid & 31;
  const int wv   = tid >> 5;
  const int lm   = lane & 15;
  const int hi   = lane >> 4;

  const int h = blockIdx.x, c = blockIdx.y, b = blockIdx.z;
  const int row0 = b * SEQ + c * CHUNK;
  const float* rbase = xconv + (size_t)row0 * CONV_DIM;

  if (tid < CHUNK) {
    dts[tid] = dt[(size_t)(row0 + tid) * NHEADS + h];
    Acs[tid] = dA[(size_t)(row0 + tid) * NHEADS + h];
  }
  if (wv == 0) {
    tdm_load_2d(&Bsm[0][0], rbase + D_INNER,           D_STATE, CHUNK, CONV_DIM, 6, 1); // 128 + 2 pad
    tdm_load_2d(&Csm[0][0], rbase + D_INNER + D_STATE, D_STATE, CHUNK, CONV_DIM, 6, 1);
    tdm_load_2d(&Xs[0][0],  rbase + h * HEADDIM,       HEADDIM, CHUNK, CONV_DIM, 5, 1); // 64 + 2 pad
    __builtin_amdgcn_s_wait_tensorcnt(0);
  }
  __syncthreads();
  if (tid == 0) {
    float s = 0.f;
    for (int l = 0; l < CHUNK; ++l) { s += Acs[l]; Acs[l] = s; }
    dAsum[(size_t)(b * NCHUNK + c) * NHEADS + h] = s;
  }
  for (int i = tid; i < CHUNK * HEADDIM; i += 256) {
    const int l = i >> 6, p = i & 63;
    Xs[l][p] *= dts[l];                                  // x * dt in place
  }
  __syncthreads();

  const int tr  = wv >> 1;
  const int tc0 = (wv & 1) * 2;

  // G = C * B^T   (64x64, K = 128)
  v8f g0 = {0.f,0.f,0.f,0.f,0.f,0.f,0.f,0.f};
  v8f g1 = g0;
  for (int k = 0; k < D_STATE; k += 4) {
    const int k0 = k + hi * 2;
    v2f a, b0, b1;
    a[0]  = Csm[tr * 16 + lm][k0];
    a[1]  = Csm[tr * 16 + lm][k0 + 1];
    b0[0] = Bsm[tc0 * 16 + lm][k0];
    b0[1] = Bsm[tc0 * 16 + lm][k0 + 1];
    b1[0] = Bsm[(tc0 + 1) * 16 + lm][k0];
    b1[1] = Bsm[(tc0 + 1) * 16 + lm][k0 + 1];
    g0 = wmma_f32(a, b0, g0);
    g1 = wmma_f32(a, b1, g1);
  }
  #pragma unroll
  for (int v = 0; v < 8; ++v) {
    const int m  = tr * 16 + v + 8 * hi;
    const int s0 = tc0 * 16 + lm, s1 = s0 + 16;
    Gs[m][s0] = (s0 <= m) ? g0[v] * __expf(Acs[m] - Acs[s0]) : 0.f;
    Gs[m][s1] = (s1 <= m) ? g1[v] * __expf(Acs[m] - Acs[s1]) : 0.f;
  }
  __syncthreads();

  // Y_diag = (G o L) @ (x*dt)   (64x64, K = 64)
  v8f y0 = {0.f,0.f,0.f,0.f,0.f,0.f,0.f,0.f};
  v8f y1 = y0;
  for (int k = 0; k < CHUNK; k += 4) {
    const int k0 = k + hi * 2;
    v2f a, b0, b1;
    a[0]  = Gs[tr * 16 + lm][k0];
    a[1]  = Gs[tr * 16 + lm][k0 + 1];
    b0[0] = Xs[k0][tc0 * 16 + lm];
    b0[1] = Xs[k0 + 1][tc0 * 16 + lm];
    b1[0] = Xs[k0][(tc0 + 1) * 16 + lm];
    b1[1] = Xs[k0 + 1][(tc0 + 1) * 16 + lm];
    y0 = wmma_f32(a, b0, y0);
    y1 = wmma_f32(a, b1, y1);
  }
  const size_t ydb = (size_t)((b * NCHUNK + c) * NHEADS + h) * (CHUNK * HEADDIM);
  #pragma unroll
  for (int v = 0; v < 8; ++v) {
    const int m = tr * 16 + v + 8 * hi;
    ydiag[ydb + m * HEADDIM + tc0 * 16 + lm]       = y0[v];
    ydiag[ydb + m * HEADDIM + (tc0 + 1) * 16 + lm] = y1[v];
  }

  // states[p][n] = sum_l (x*dt)[l][p] * decay[l] * B[l][n]   (64x128, K = 64)
  const int tr2 = wv >> 1;
  const int tcB = (wv & 1) * 4;
  v8f sacc[4];
  #pragma unroll
  for (int j = 0; j < 4; ++j) sacc[j] = (v8f){0.f,0.f,0.f,0.f,0.f,0.f,0.f,0.f};
  const float atot = Acs[CHUNK - 1];
  for (int k = 0; k < CHUNK; k += 4) {
    const int k0 = k + hi * 2, k1 = k0 + 1;
    const float d0 = __expf(atot - Acs[k0]);
    const float d1 = __expf(atot - Acs[k1]);
    v2f a;
    a[0] = Xs[k0][tr2 * 16 + lm] * d0;
    a[1] = Xs[k1][tr2 * 16 + lm] * d1;
    #pragma unroll
    for (int j = 0; j < 4; ++j) {
      v2f bb;
      bb[0] = Bsm[k0][(tcB + j) * 16 + lm];
      bb[1] = Bsm[k1][(tcB + j) * 16 + lm];
      sacc[j] = wmma_f32(a, bb, sacc[j]);
    }
  }
  const size_t stb = (size_t)((b * NCHUNK + c) * NHEADS + h) * (HEADDIM * D_STATE);
  #pragma unroll
  for (int j = 0; j < 4; ++j)
    #pragma unroll
    for (int v = 0; v < 8; ++v) {
      const int p = tr2 * 16 + v + 8 * hi;
      const int n = (tcB + j) * 16 + lm;
      states[stb + p * D_STATE + n] = sacc[j][v];
    }
}

// ---------------------------------------------------------------------------
// Inter-chunk state scan: P_0 = 0; P_{c+1} = exp(Asum_c)*P_c + state_c
// ---------------------------------------------------------------------------
__global__ void scan_kernel(const float* __restrict__ states, const float* __restrict__ dAsum,
                            float* __restrict__ sprev) {
  const int el = blockIdx.x * 256 + threadIdx.x;
  const int h = blockIdx.y, b = blockIdx.z;
  float carry = 0.f;
  for (int c = 0; c < NCHUNK; ++c) {
    const size_t off = (size_t)((b * NCHUNK + c) * NHEADS + h) * (HEADDIM * D_STATE) + el;
    sprev[off] = carry;
    carry = carry * __expf(dAsum[(size_t)(b * NCHUNK + c) * NHEADS + h]) + states[off];
  }
}

// ---------------------------------------------------------------------------
// Y_off = exp(Acs) * (C @ sprev^T);  Y = Y_diag + Y_off + D*x
// ---------------------------------------------------------------------------
__global__ void combine_kernel(const float* __restrict__ xconv, const float* __restrict__ dA,
                               const float* __restrict__ sprev, const float* __restrict__ ydiag,
                               const float* __restrict__ D_param, float* __restrict__ yfull) {
  __shared__ float Csm[CHUNK][130];
  __shared__ float Ssm[HEADDIM][130];
  __shared__ float Acs[CHUNK];

  const int tid  = threadIdx.x;
  const int lane = tid & 31;
  const int wv   = tid >> 5;
  const int lm   = lane & 15;
  const int hi   = lane >> 4;

  const int h = blockIdx.x, c = blockIdx.y, b = blockIdx.z;
  const int row0 = b * SEQ + c * CHUNK;
  const size_t spb = (size_t)((b * NCHUNK + c) * NHEADS + h) * (HEADDIM * D_STATE);

  if (tid < CHUNK) Acs[tid] = dA[(size_t)(row0 + tid) * NHEADS + h];
  if (wv == 0) {
    tdm_load_2d(&Csm[0][0], xconv + (size_t)row0 * CONV_DIM + D_INNER + D_STATE,
                D_STATE, CHUNK, CONV_DIM, 6, 1);
    tdm_load_2d(&Ssm[0][0], sprev + spb, D_STATE, HEADDIM, D_STATE, 6, 1);
    __builtin_amdgcn_s_wait_tensorcnt(0);
  }
  __syncthreads();
  if (tid == 0) {
    float s = 0.f;
    for (int l = 0; l < CHUNK; ++l) { s += Acs[l]; Acs[l] = s; }
  }
  __syncthreads();

  const int tr  = wv >> 1;
  const int tc0 = (wv & 1) * 2;

  v8f y0 = {0.f,0.f,0.f,0.f,0.f,0.f,0.f,0.f};
  v8f y1 = y0;
  for (int k = 0; k < D_STATE; k += 4) {
    const int k0 = k + hi * 2;
    v2f a, b0, b1;
    a[0]  = Csm[tr * 16 + lm][k0];
    a[1]  = Csm[tr * 16 + lm][k0 + 1];
    b0[0] = Ssm[tc0 * 16 + lm][k0];
    b0[1] = Ssm[tc0 * 16 + lm][k0 + 1];
    b1[0] = Ssm[(tc0 + 1) * 16 + lm][k0];
    b1[1] = Ssm[(tc0 + 1) * 16 + lm][k0 + 1];
    y0 = wmma_f32(a, b0, y0);
    y1 = wmma_f32(a, b1, y1);
  }

  const float Dh = D_param[h];
  const size_t ydb = (size_t)((b * NCHUNK + c) * NHEADS + h) * (CHUNK * HEADDIM);
  #pragma unroll
  for (int v = 0; v < 8; ++v) {
    const int m = tr * 16 + v + 8 * hi;
    const float ea = __expf(Acs[m]);
    #pragma unroll
    for (int s = 0; s < 2; ++s) {
      const int p = (tc0 + s) * 16 + lm;
      const float yo = (s == 0 ? y0[v] : y1[v]) * ea;
      const float yd = ydiag[ydb + m * HEADDIM + p];
      const float xr = xconv[(size_t)(row0 + m) * CONV_DIM + h * HEADDIM + p];
      yfull[(size_t)(row0 + m) * D_INNER + h * HEADDIM + p] = yd + yo + Dh * xr;
    }
  }
}

// ---------------------------------------------------------------------------
__global__ void gatenorm_kernel(const float* __restrict__ zx, const float* __restrict__ yfull,
                                const float* __restrict__ norm_w, float* __restrict__ ynorm) {
  __shared__ float red[256];
  const int row = blockIdx.x, tid = threadIdx.x;
  const float* zr = zx + (size_t)row * D_IN_PROJ;
  const float* yr = yfull + (size_t)row * D_INNER;
  float loc[8];
  float ss = 0.f;
  #pragma unroll
  for (int j = 0; j < 8; ++j) {
    const int d = tid + j * 256;
    const float z = zr[d];
    const float v = yr[d] * z * sigmoidf_(z);
    loc[j] = v;
    ss += v * v;
  }
  red[tid] = ss;
  __syncthreads();
  for (int s = 128; s > 0; s >>= 1) {
    if (tid < s) red[tid] += red[tid + s];
    __syncthreads();
  }
  const float rms = rsqrtf(red[0] / (float)D_INNER + 1e-5f);
  #pragma unroll
  for (int j = 0; j < 8; ++j) {
    const int d = tid + j * 256;
    ynorm[(size_t)row * D_INNER + d] = loc[j] * rms * norm_w[d];
  }
}

// ---------------------------------------------------------------------------
extern "C" void kernel_launch(void* const* d_in, const int* in_sizes, int n_in,
                              void* d_out, int out_size, void* d_ws, size_t ws_size,
                              hipStream_t stream) {
  const float* u          = (const float*)d_in[0];
  const float* in_proj_w  = (const float*)d_in[1];
  const float* conv_w     = (const float*)d_in[2];
  const float* conv_b     = (const float*)d_in[3];
  const float* dt_bias    = (const float*)d_in[4];
  const float* A_log      = (const float*)d_in[5];
  const float* D_param    = (const float*)d_in[6];
  const float* norm_w     = (const float*)d_in[7];
  const float* out_proj_w = (const float*)d_in[8];
  float* out = (float*)d_out;

  float* ws = (float*)d_ws;
  size_t o = 0;
  float* zx     = ws + o; o += (size_t)BL * D_IN_PROJ;
  float* xconv  = ws + o; o += (size_t)BL * CONV_DIM;
  float* dtb    = ws + o; o += (size_t)BL * NHEADS;
  float* dAb    = ws + o; o += (size_t)BL * NHEADS;
  float* dAsum  = ws + o; o += (size_t)BATCH * NCHUNK * NHEADS;
  float* states = ws + o; o += (size_t)BATCH * NCHUNK * NHEADS * HEADDIM * D_STATE;
  float* sprev  = ws + o; o += (size_t)BATCH * NCHUNK * NHEADS * HEADDIM * D_STATE;
  float* ydiag  = ws + o; o += (size_t)BATCH * NCHUNK * NHEADS * CHUNK * HEADDIM;
  float* yfull  = ws + o; o += (size_t)BL * D_INNER;
  float* ynorm  = ws + o; o += (size_t)BL * D_INNER;

  gemm_f32<<<dim3(D_IN_PROJ / 32, BL / 256), 256, 0, stream>>>(u, in_proj_w, zx, BL, D_IN_PROJ, D_MODEL);
  dt_kernel<<<(BL * NHEADS) / 256, 256, 0, stream>>>(zx, dt_bias, A_log, dtb, dAb);
  conv_kernel<<<(BL * CONV_DIM + 255) / 256, 256, 0, stream>>>(zx, conv_w, conv_b, xconv);
  chunk_kernel<<<dim3(NHEADS, NCHUNK, BATCH), 256, 0, stream>>>(xconv, dtb, dAb, ydiag, states, dAsum);
  scan_kernel<<<dim3((HEADDIM * D_STATE) / 256, NHEADS, BATCH), 256, 0, stream>>>(states, dAsum, sprev);
  combine_kernel<<<dim3(NHEADS, NCHUNK, BATCH), 256, 0, stream>>>(xconv, dAb, sprev, ydiag, D_param, yfull);
  gatenorm_kernel<<<BL, 256, 0, stream>>>(zx, yfull, norm_w, ynorm);
  gemm_f32<<<dim3(D_MODEL / 32, BL / 256), 256, 0, stream>>>(ynorm, out_proj_w, out, BL, D_MODEL, D_INNER);
}